// LogSparseAttention_79843442032893
// MI455X (gfx1250) — compile-verified
//
#include <hip/hip_runtime.h>

// LogSparse attention, CDNA5 (gfx1250), wave32.
// out[b,i,h,:] = softmax_j( Q[b,i,h]·K[b,j,h] / 8 ) · V[b,j,h]
// where j ∈ {i} ∪ {i - 2^k : 2^k <= i}. At most 12 keys per query.
//
// One wave handles a 16-query tile of one (b,h). Scores for each offset o are
// the diagonal of Qtile(16x64) · Ktile(64x16), computed exactly in fp32 with a
// chain of 16 V_WMMA_F32_16X16X4_F32. Softmax + P·V are per-lane VALU, with
// the V phase using float4 (global_load_b128) on contiguous half-rows.

typedef __attribute__((ext_vector_type(2))) float v2f;
typedef __attribute__((ext_vector_type(4))) float v4f;
typedef __attribute__((ext_vector_type(8))) float v8f;

#define B_    4
#define L_    2048
#define H_    8
#define D_    64
#define NOFF  12
#define NTILE (L_ / 16)      // 128 tiles per (b,h)

__global__ __launch_bounds__(256)
void logsparse_attn_kernel(const float* __restrict__ Q,
                           const float* __restrict__ K,
                           const float* __restrict__ V,
                           float* __restrict__ O) {
    const int lane = threadIdx.x & 31;
    const int wid  = (blockIdx.x * 256 + threadIdx.x) >> 5;   // global wave id
    const int tile = wid & (NTILE - 1);
    const int bh   = wid >> 7;                // 0..31
    const int h    = bh & (H_ - 1);
    const int b    = bh >> 3;

    const int r    = lane & 15;               // row/column index within tile
    const int hi   = lane >> 4;               // lane half
    const int i0   = tile << 4;
    const int qrow = i0 + r;

    const int rowstride = H_ * D_;                       // 512 floats
    const int bhbase    = b * L_ * rowstride + h * D_;   // + row*rowstride + d

    // ---- WMMA operand dim pattern: this lane supplies K-dims {4c+2*hi, +1}.
    const int dbase = hi * 2;

    // ---- Load A operand: Q tile.
    v2f a[16];
    {
        const float* qp = Q + bhbase + qrow * rowstride + dbase;
        #pragma unroll
        for (int c = 0; c < 16; ++c)
            a[c] = *(const v2f*)(qp + 4 * c);
    }

    const int offs[NOFF] = {0, 1, 2, 4, 8, 16, 32, 64, 128, 256, 512, 1024};
    float s_arr[NOFF];

    // ---- Scores per offset via WMMA, keep only the diagonal.
    #pragma unroll
    for (int oi = 0; oi < NOFF; ++oi) {
        const int o    = offs[oi];
        const int krow = qrow - o;                         // key row for column r
        const int krc  = krow < 0 ? 0 : krow;              // clamp for the load; masked later
        const float* kp = K + bhbase + krc * rowstride + dbase;

        v8f acc = {0.f, 0.f, 0.f, 0.f, 0.f, 0.f, 0.f, 0.f};
        #pragma unroll
        for (int c = 0; c < 16; ++c) {
            v2f bm = *(const v2f*)(kp + 4 * c);            // B column r, K-dims 4c+2*hi..+1
            acc = __builtin_amdgcn_wmma_f32_16x16x4_f32(
                /*neg_a=*/false, a[c], /*neg_b=*/false, bm,
                /*c_mod=*/(short)0, acc, /*reuse_a=*/false, /*reuse_b=*/false);
        }

        // Diagonal S[m][m]: row m lives at lane (m<8 ? m : m+16), VGPR m&7.
        // Every lane selects acc[lane&7]; lane with row r reads lane r+((r&8)<<1).
        const int sel = lane & 7;
        float x = acc[0];
        x = (sel == 1) ? acc[1] : x;
        x = (sel == 2) ? acc[2] : x;
        x = (sel == 3) ? acc[3] : x;
        x = (sel == 4) ? acc[4] : x;
        x = (sel == 5) ? acc[5] : x;
        x = (sel == 6) ? acc[6] : x;
        x = (sel == 7) ? acc[7] : x;
        const int src = r + ((r & 8) << 1);
        float s = __shfl(x, src, 32);

        s_arr[oi] = (krow >= 0) ? s * 0.125f : -__builtin_inff();   // scale = 1/sqrt(64)
    }

    // ---- Per-row softmax over <=12 scores (o=0 always valid -> max is finite).
    float m = s_arr[0];
    #pragma unroll
    for (int oi = 1; oi < NOFF; ++oi) m = fmaxf(m, s_arr[oi]);
    float p[NOFF];
    float sum = 0.f;
    #pragma unroll
    for (int oi = 0; oi < NOFF; ++oi) {
        p[oi] = __expf(s_arr[oi] - m);     // exp(-inf) = 0 for masked entries
        sum += p[oi];
    }
    const float inv = 1.0f / sum;

    // ---- out[r][d] = sum_o p_o * V[qrow-o][d].
    // V phase is layout-free: lane handles contiguous dims [hi*32, hi*32+32)
    // of row r, so we can use 16B (b128) loads/stores.
    const int vdbase = hi * 32;
    v4f oacc[8];
    #pragma unroll
    for (int c = 0; c < 8; ++c) { oacc[c].x = 0.f; oacc[c].y = 0.f; oacc[c].z = 0.f; oacc[c].w = 0.f; }

    #pragma unroll
    for (int oi = 0; oi < NOFF; ++oi) {
        int vrow = qrow - offs[oi];
        if (vrow < 0) vrow = 0;            // weight is 0 for masked rows
        const float w = p[oi] * inv;
        const float* vp = V + bhbase + vrow * rowstride + vdbase;
        #pragma unroll
        for (int c = 0; c < 8; ++c) {
            v4f vv = *(const v4f*)(vp + 4 * c);
            oacc[c].x += w * vv.x;
            oacc[c].y += w * vv.y;
            oacc[c].z += w * vv.z;
            oacc[c].w += w * vv.w;
        }
    }

    float* op = O + bhbase + qrow * rowstride + vdbase;
    #pragma unroll
    for (int c = 0; c < 8; ++c)
        *(v4f*)(op + 4 * c) = oacc[c];
}

extern "C" void kernel_launch(void* const* d_in, const int* in_sizes, int n_in,
                              void* d_out, int out_size, void* d_ws, size_t ws_size,
                              hipStream_t stream) {
    const float* Q = (const float*)d_in[0];
    const float* K = (const float*)d_in[1];
    const float* V = (const float*)d_in[2];
    float* O = (float*)d_out;

    // B*H*(L/16) = 4096 waves; 8 waves (256 threads) per block -> 512 blocks.
    dim3 grid(B_ * H_ * NTILE / 8);
    dim3 block(256);
    hipLaunchKernelGGL(logsparse_attn_kernel, grid, block, 0, stream, Q, K, V, O);
}